// GraphEncoder_26946624815681
// MI455X (gfx1250) — compile-verified
//
#include <hip/hip_runtime.h>

typedef __attribute__((ext_vector_type(2))) float v2f;
typedef __attribute__((ext_vector_type(8))) float v8f;

#define HIDDEN 256

// ---------------------------------------------------------------- utilities
__global__ void zero_kernel(float* __restrict__ p, long n) {
    long i = (long)blockIdx.x * blockDim.x + threadIdx.x;
    long stride = (long)gridDim.x * blockDim.x;
    for (; i < n; i += stride) p[i] = 0.0f;
}

__global__ void degree_kernel(const int* __restrict__ dst, float* __restrict__ cnt, int E) {
    int e = blockIdx.x * blockDim.x + threadIdx.x;
    if (e < E) atomicAdd(&cnt[dst[e]], 1.0f);
}

// One wave32 handles a 128-float chunk of one edge (float4 per lane).
// src/dst are wave-uniform -> scalar loads; adds lower to global_atomic_add_f32.
template <int D>
__global__ void scatter_kernel(const float* __restrict__ feat, const int* __restrict__ src,
                               const int* __restrict__ dst, float* __restrict__ agg, int E) {
    constexpr int CHUNKS = D / 128;
    int gw = blockIdx.x * (blockDim.x >> 5) + (threadIdx.x >> 5);
    int lane = threadIdx.x & 31;
    int e = gw / CHUNKS;
    int part = gw - e * CHUNKS;
    if (e >= E) return;
    int s = src[e];
    int d = dst[e];
    int off = part * 128 + lane * 4;
    const float4 v = *(const float4*)(feat + (long)s * D + off);
    float* a = agg + (long)d * D + off;
    atomicAdd(a + 0, v.x);
    atomicAdd(a + 1, v.y);
    atomicAdd(a + 2, v.z);
    atomicAdd(a + 3, v.w);
}

// -------------------------------------------------- fused SAGE + LN + ReLU
// Block: 512 threads = 16 waves; computes a 16-node x 256-col output tile.
//   out_row = LN( (A1/deg)@W1 + A2@W2 + bias ) * gamma + beta, ReLU,
//   plus (ADD_BASE) residual X@Wres + bres added after activation.
// All matmuls use V_WMMA_F32_16X16X4_F32.
template <int KMAIN, bool ADD_BASE>
__global__ __launch_bounds__(512) void sage_ln_kernel(
    const float* __restrict__ A1, const float* __restrict__ cnt, const float* __restrict__ A2,
    const float* __restrict__ W1, const float* __restrict__ W2, const float* __restrict__ bias,
    const float* __restrict__ gamma, const float* __restrict__ beta, const float* __restrict__ X,
    const float* __restrict__ Wres, const float* __restrict__ bres, float* __restrict__ out) {
    constexpr int LDA = KMAIN + 4;        // pad to avoid LDS bank conflicts
    constexpr int A_ELEMS = 16 * LDA;
    constexpr int X_ELEMS = ADD_BASE ? 16 * 132 : 0;
    constexpr int NEED_OUT = ADD_BASE ? 8192 : 4096;  // sOut (+ sBase) overlay
    constexpr int SMEM = (2 * A_ELEMS + X_ELEMS) > NEED_OUT ? (2 * A_ELEMS + X_ELEMS) : NEED_OUT;
    __shared__ float smem[SMEM];
    __shared__ float sInv[16];

    float* sA1 = smem;
    float* sA2 = smem + A_ELEMS;
    float* sX = smem + 2 * A_ELEMS;
    float* sOut = smem;          // aliases A tiles after barrier
    float* sBase = smem + 4096;  // aliases A tiles after barrier (ADD_BASE only)

    const int tid = threadIdx.x;
    const int node0 = blockIdx.x * 16;

    if (tid < 16) {
        float c = cnt[node0 + tid];
        sInv[tid] = 1.0f / fmaxf(c, 1.0f);
    }
    __syncthreads();

    // cooperative tile loads (mean-normalize agg on the way in)
    for (int i = tid; i < 16 * KMAIN; i += 512) {
        int r = i / KMAIN, k = i - r * KMAIN;
        sA1[r * LDA + k] = A1[(long)(node0 + r) * KMAIN + k] * sInv[r];
        sA2[r * LDA + k] = A2[(long)(node0 + r) * KMAIN + k];
    }
    if (ADD_BASE) {
        for (int i = tid; i < 16 * 128; i += 512) {
            int r = i >> 7, k = i & 127;
            sX[r * 132 + k] = X[(long)(node0 + r) * 128 + k];
        }
    }
    __syncthreads();

    const int wave = tid >> 5;
    const int lane = tid & 31;
    const int nloc = lane & 15;          // A-row / B-col within tile
    const int khalf = (lane >> 4) * 2;   // K sub-pair selector
    const int n0 = wave * 16;
    const int col = n0 + nloc;

    v8f c = {0.f, 0.f, 0.f, 0.f, 0.f, 0.f, 0.f, 0.f};
    for (int k = 0; k < KMAIN; k += 4) {
        v2f a, b;
        a.x = sA1[nloc * LDA + k + khalf];
        a.y = sA1[nloc * LDA + k + khalf + 1];
        b.x = W1[(k + khalf) * HIDDEN + col];
        b.y = W1[(k + khalf + 1) * HIDDEN + col];
        c = __builtin_amdgcn_wmma_f32_16x16x4_f32(false, a, false, b, (short)0, c, false, false);
    }
    for (int k = 0; k < KMAIN; k += 4) {
        v2f a, b;
        a.x = sA2[nloc * LDA + k + khalf];
        a.y = sA2[nloc * LDA + k + khalf + 1];
        b.x = W2[(k + khalf) * HIDDEN + col];
        b.y = W2[(k + khalf + 1) * HIDDEN + col];
        c = __builtin_amdgcn_wmma_f32_16x16x4_f32(false, a, false, b, (short)0, c, false, false);
    }

    v8f cb = {0.f, 0.f, 0.f, 0.f, 0.f, 0.f, 0.f, 0.f};
    if (ADD_BASE) {
        for (int k = 0; k < 128; k += 4) {
            v2f a, b;
            a.x = sX[nloc * 132 + k + khalf];
            a.y = sX[nloc * 132 + k + khalf + 1];
            b.x = Wres[(k + khalf) * HIDDEN + col];
            b.y = Wres[(k + khalf + 1) * HIDDEN + col];
            cb = __builtin_amdgcn_wmma_f32_16x16x4_f32(false, a, false, b, (short)0, cb, false,
                                                       false);
        }
    }

    __syncthreads();  // everyone done reading A tiles; safe to overlay sOut/sBase

    // spill accumulators (+bias) to LDS in row-major form
    const float bn = bias[col];
#pragma unroll
    for (int v = 0; v < 8; ++v) {
        int row = v + 8 * (lane >> 4);
        sOut[row * HIDDEN + col] = c[v] + bn;
    }
    if (ADD_BASE) {
        const float brn = bres[col];
#pragma unroll
        for (int v = 0; v < 8; ++v) {
            int row = v + 8 * (lane >> 4);
            sBase[row * HIDDEN + col] = cb[v] + brn;
        }
    }
    __syncthreads();

    // LayerNorm + ReLU (+base) : wave w owns row w (32 lanes x 8 cols)
    {
        const int row = wave;
        float s = 0.f, ss = 0.f;
        float vals[8];
#pragma unroll
        for (int j = 0; j < 8; ++j) {
            float x = sOut[row * HIDDEN + lane + 32 * j];
            vals[j] = x;
            s += x;
            ss += x * x;
        }
#pragma unroll
        for (int m = 16; m >= 1; m >>= 1) {
            s += __shfl_xor(s, m, 32);
            ss += __shfl_xor(ss, m, 32);
        }
        const float mu = s * (1.0f / HIDDEN);
        const float var = ss * (1.0f / HIDDEN) - mu * mu;
        const float rsig = rsqrtf(var + 1e-5f);
        float* op = out + (long)(node0 + row) * HIDDEN;
#pragma unroll
        for (int j = 0; j < 8; ++j) {
            int cc = lane + 32 * j;
            float y = (vals[j] - mu) * rsig * gamma[cc] + beta[cc];
            y = fmaxf(y, 0.0f);
            if (ADD_BASE) y += sBase[row * HIDDEN + cc];
            op[cc] = y;
        }
    }
}

// ----------------------------------------------------------------- launcher
extern "C" void kernel_launch(void* const* d_in, const int* in_sizes, int n_in, void* d_out,
                              int out_size, void* d_ws, size_t ws_size, hipStream_t stream) {
    const float* x = (const float*)d_in[0];
    const int* edge = (const int*)d_in[1];
    const float* Wl1 = (const float*)d_in[2];
    const float* bl1 = (const float*)d_in[3];
    const float* Wr1 = (const float*)d_in[4];
    const float* g1 = (const float*)d_in[5];
    const float* be1 = (const float*)d_in[6];
    const float* Wl2 = (const float*)d_in[7];
    const float* bl2 = (const float*)d_in[8];
    const float* Wr2 = (const float*)d_in[9];
    const float* g2 = (const float*)d_in[10];
    const float* be2 = (const float*)d_in[11];
    const float* Wres = (const float*)d_in[12];
    const float* bres = (const float*)d_in[13];

    const int N = in_sizes[0] / 128;
    const int E = in_sizes[1] / 2;
    const int* src = edge;
    const int* dst = edge + E;

    // workspace layout (floats)
    float* ws = (float*)d_ws;
    float* cnt = ws;                          // N
    float* agg1 = ws + N;                     // N*128   (contiguous after cnt)
    float* h1 = agg1 + (long)N * 128;         // N*256
    float* agg2 = h1 + (long)N * 256;         // N*256
    float* outp = (float*)d_out;

    // 1) clear cnt+agg1 and agg2
    zero_kernel<<<4096, 256, 0, stream>>>(cnt, (long)N + (long)N * 128);
    zero_kernel<<<4096, 256, 0, stream>>>(agg2, (long)N * 256);

    // 2) in-degrees
    degree_kernel<<<(E + 255) / 256, 256, 0, stream>>>(dst, cnt, E);

    // 3) layer-1 scatter-add of x into agg1 (d=128 -> 1 wave/edge)
    {
        long waves = (long)E;
        int blocks = (int)((waves + 7) / 8);
        scatter_kernel<128><<<blocks, 256, 0, stream>>>(x, src, dst, agg1, E);
    }

    // 4) layer-1 fused SAGE + LN + ReLU  -> h1
    sage_ln_kernel<128, false><<<N / 16, 512, 0, stream>>>(agg1, cnt, x, Wl1, Wr1, bl1, g1, be1,
                                                           nullptr, nullptr, nullptr, h1);

    // 5) layer-2 scatter-add of h1 into agg2 (d=256 -> 2 waves/edge)
    {
        long waves = (long)E * 2;
        int blocks = (int)((waves + 7) / 8);
        scatter_kernel<256><<<blocks, 256, 0, stream>>>(h1, src, dst, agg2, E);
    }

    // 6) layer-2 fused SAGE + LN + ReLU + residual(x@Wres+bres) -> d_out
    sage_ln_kernel<256, true><<<N / 16, 512, 0, stream>>>(agg2, cnt, h1, Wl2, Wr2, bl2, g2, be2, x,
                                                          Wres, bres, outp);
}